// ParallelQuantumModel_4398046511620
// MI455X (gfx1250) — compile-verified
//
#include <hip/hip_runtime.h>
#include <hip/hip_bf16.h>
#include <math.h>

#define N_QUBITS 6
#define NAMP 64
#define KDIM 2048
#define KHALF_LEN 1024
#define NOUT1 64

typedef __attribute__((ext_vector_type(2))) float v2f;
typedef __attribute__((ext_vector_type(8))) float v8f;

// ---------------------------------------------------------------------------
// Kernel A: partial[ks] = xc[:, ks*1024:(ks+1)*1024] @ W1[:, same]^T
// One wave computes a 32(M) x 64(N) partial tile over half of K using
// V_WMMA_F32_16X16X4_F32 (8 accumulators, 6 b64 loads per 8 WMMAs).
// Block = 256 threads = 8 waves = 4 M-chunks x 2 K-slices -> 128 rows/block.
// Bias + ReLU are deferred to the combine kernel (partials must sum first).
// ---------------------------------------------------------------------------
__global__ __launch_bounds__(256) void gemm_partial_wmma(
    const float* __restrict__ xc, const float* __restrict__ W1,
    float* __restrict__ p_out /* [2][B][64] */, int B)
{
    const int lane  = threadIdx.x & 31;
    const int wave  = threadIdx.x >> 5;
    const int l15   = lane & 15;
    const int khalf = (lane >> 4) * 2;            // K sub-offset per half-wave

    const int row0   = (blockIdx.x * 4 + (wave >> 1)) * 32;  // 32 M-rows/wave
    const int kslice = wave & 1;                              // which K half
    const int kbase  = kslice * KHALF_LEN;

    // A fragments: rows row0+l15 (lower 16) and row0+16+l15 (upper 16)
    const float* arow0 = xc + (size_t)(row0 + l15) * KDIM + kbase + khalf;
    const float* arow1 = arow0 + (size_t)16 * KDIM;
    // B fragments: W1 rows n = t*16 + l15
    const float* brow0 = W1 + (size_t)( 0 + l15) * KDIM + kbase + khalf;
    const float* brow1 = W1 + (size_t)(16 + l15) * KDIM + kbase + khalf;
    const float* brow2 = W1 + (size_t)(32 + l15) * KDIM + kbase + khalf;
    const float* brow3 = W1 + (size_t)(48 + l15) * KDIM + kbase + khalf;

    v8f accA0 = {}, accA1 = {}, accA2 = {}, accA3 = {};
    v8f accB0 = {}, accB1 = {}, accB2 = {}, accB3 = {};

#pragma unroll 8
    for (int k = 0; k < KHALF_LEN; k += 4) {
        v2f a0 = *(const v2f*)(arow0 + k);
        v2f a1 = *(const v2f*)(arow1 + k);
        v2f b0 = *(const v2f*)(brow0 + k);
        v2f b1 = *(const v2f*)(brow1 + k);
        v2f b2 = *(const v2f*)(brow2 + k);
        v2f b3 = *(const v2f*)(brow3 + k);
        accA0 = __builtin_amdgcn_wmma_f32_16x16x4_f32(false, a0, false, b0, (short)0, accA0, false, false);
        accA1 = __builtin_amdgcn_wmma_f32_16x16x4_f32(false, a0, false, b1, (short)0, accA1, false, false);
        accA2 = __builtin_amdgcn_wmma_f32_16x16x4_f32(false, a0, false, b2, (short)0, accA2, false, false);
        accA3 = __builtin_amdgcn_wmma_f32_16x16x4_f32(false, a0, false, b3, (short)0, accA3, false, false);
        accB0 = __builtin_amdgcn_wmma_f32_16x16x4_f32(false, a1, false, b0, (short)0, accB0, false, false);
        accB1 = __builtin_amdgcn_wmma_f32_16x16x4_f32(false, a1, false, b1, (short)0, accB1, false, false);
        accB2 = __builtin_amdgcn_wmma_f32_16x16x4_f32(false, a1, false, b2, (short)0, accB2, false, false);
        accB3 = __builtin_amdgcn_wmma_f32_16x16x4_f32(false, a1, false, b3, (short)0, accB3, false, false);
    }

    // C/D layout: VGPR r -> M = r (lanes 0-15) or M = r+8 (lanes 16-31), N=l15
    float* pbase = p_out + (size_t)kslice * B * NOUT1;
    const int moff = (lane >> 4) * 8;
#pragma unroll
    for (int r = 0; r < 8; ++r) {
        const size_t mA = (size_t)(row0 + moff + r) * NOUT1;
        const size_t mB = (size_t)(row0 + 16 + moff + r) * NOUT1;
        pbase[mA +  0 + l15] = accA0[r];
        pbase[mA + 16 + l15] = accA1[r];
        pbase[mA + 32 + l15] = accA2[r];
        pbase[mA + 48 + l15] = accA3[r];
        pbase[mB +  0 + l15] = accB0[r];
        pbase[mB + 16 + l15] = accB1[r];
        pbase[mB + 32 + l15] = accB2[r];
        pbase[mB + 48 + l15] = accB3[r];
    }
}

// ---------------------------------------------------------------------------
// Kernel B: quantum circuit + q = z @ Wq^T + bq.  One sample per lane; the
// 64-amplitude complex state lives fully in VGPRs (all loops unrolled).
// Qubit w corresponds to bit (5-w) of the flat amplitude index.
// ---------------------------------------------------------------------------
__global__ __launch_bounds__(128) void quantum_kernel(
    const float* __restrict__ xq, const float* __restrict__ qw,
    const float* __restrict__ Wq, const float* __restrict__ bq,
    float* __restrict__ q_out, int B)
{
    const int b = blockIdx.x * blockDim.x + threadIdx.x;
    if (b >= B) return;

    float sr[NAMP], si[NAMP];

    // Product-state init: amp[i] = prod_w (bit ? sin(x/2) : cos(x/2))
    float cs[N_QUBITS], sn[N_QUBITS];
#pragma unroll
    for (int w = 0; w < N_QUBITS; ++w) {
        float h = 0.5f * xq[(size_t)b * N_QUBITS + w];
        sincosf(h, &sn[w], &cs[w]);
    }
#pragma unroll
    for (int i = 0; i < NAMP; ++i) {
        float a = 1.f;
#pragma unroll
        for (int w = 0; w < N_QUBITS; ++w)
            a *= ((i >> (5 - w)) & 1) ? sn[w] : cs[w];
        sr[i] = a; si[i] = 0.f;
    }

#pragma unroll
    for (int l = 0; l < 3; ++l) {
        // single-qubit Rot(phi, theta, omega) gates
#pragma unroll
        for (int w = 0; w < N_QUBITS; ++w) {
            const float phi = qw[((l * N_QUBITS + w) * 3) + 0];
            const float th  = qw[((l * N_QUBITS + w) * 3) + 1];
            const float om  = qw[((l * N_QUBITS + w) * 3) + 2];
            float ct, st; sincosf(0.5f * th, &st, &ct);
            float cp, sp; sincosf(0.5f * (phi + om), &sp, &cp);
            float cm, sm; sincosf(0.5f * (phi - om), &sm, &cm);
            // U00 = e^{-i(phi+om)/2} c; U01 = -e^{+i(phi-om)/2} s
            // U10 = e^{-i(phi-om)/2} s; U11 = e^{+i(phi+om)/2} c
            const float u00r =  cp * ct, u00i = -sp * ct;
            const float u01r = -cm * st, u01i = -sm * st;
            const float u10r =  cm * st, u10i = -sm * st;
            const float u11r =  cp * ct, u11i =  sp * ct;
            const int bit = 1 << (5 - w);
#pragma unroll
            for (int i = 0; i < NAMP; ++i) {
                if (i & bit) continue;
                const int j = i | bit;
                const float a0r = sr[i], a0i = si[i];
                const float a1r = sr[j], a1i = si[j];
                sr[i] = u00r * a0r - u00i * a0i + u01r * a1r - u01i * a1i;
                si[i] = u00r * a0i + u00i * a0r + u01r * a1i + u01i * a1r;
                sr[j] = u10r * a0r - u10i * a0i + u11r * a1r - u11i * a1i;
                si[j] = u10r * a0i + u10i * a0r + u11r * a1i + u11i * a1r;
            }
        }
        // CNOT ring: ctrl=w, tgt=(w+r)%6, r = l%5+1
        const int r = (l % (N_QUBITS - 1)) + 1;
#pragma unroll
        for (int w = 0; w < N_QUBITS; ++w) {
            const int cb = 1 << (5 - w);
            const int tb = 1 << (5 - ((w + r) % N_QUBITS));
#pragma unroll
            for (int i = 0; i < NAMP; ++i) {
                if ((i & cb) && !(i & tb)) {
                    const int j = i | tb;
                    float tr = sr[i]; sr[i] = sr[j]; sr[j] = tr;
                    float ti = si[i]; si[i] = si[j]; si[j] = ti;
                }
            }
        }
    }

    // z expectations
    float zs[N_QUBITS];
#pragma unroll
    for (int w = 0; w < N_QUBITS; ++w) zs[w] = 0.f;
#pragma unroll
    for (int i = 0; i < NAMP; ++i) {
        const float p = sr[i] * sr[i] + si[i] * si[i];
#pragma unroll
        for (int w = 0; w < N_QUBITS; ++w)
            zs[w] += ((i >> (5 - w)) & 1) ? -p : p;
    }

    // q = z @ Wq^T + bq   (6 -> 16)
#pragma unroll
    for (int j = 0; j < 16; ++j) {
        float acc = bq[j];
#pragma unroll
        for (int w = 0; w < N_QUBITS; ++w)
            acc += zs[w] * Wq[j * N_QUBITS + w];
        q_out[(size_t)b * 16 + j] = acc;
    }
}

// ---------------------------------------------------------------------------
// Kernel C: c = relu(p0 + p1 + b1); out = [c, q] @ Wf^T + bf   (80 -> 3)
// ---------------------------------------------------------------------------
__global__ __launch_bounds__(256) void combine_kernel(
    const float* __restrict__ p0, const float* __restrict__ p1,
    const float* __restrict__ b1, const float* __restrict__ q,
    const float* __restrict__ Wf, const float* __restrict__ bf,
    float* __restrict__ out, int B)
{
    const int b = blockIdx.x * blockDim.x + threadIdx.x;
    if (b >= B) return;

    float acc0 = bf[0], acc1 = bf[1], acc2 = bf[2];
#pragma unroll 16
    for (int i = 0; i < 64; ++i) {
        float c = p0[(size_t)b * 64 + i] + p1[(size_t)b * 64 + i] + b1[i];
        c = c > 0.f ? c : 0.f;
        acc0 += c * Wf[0 * 80 + i];
        acc1 += c * Wf[1 * 80 + i];
        acc2 += c * Wf[2 * 80 + i];
    }
#pragma unroll
    for (int i = 0; i < 16; ++i) {
        const float qv = q[(size_t)b * 16 + i];
        acc0 += qv * Wf[0 * 80 + 64 + i];
        acc1 += qv * Wf[1 * 80 + 64 + i];
        acc2 += qv * Wf[2 * 80 + 64 + i];
    }
    out[(size_t)b * 3 + 0] = acc0;
    out[(size_t)b * 3 + 1] = acc1;
    out[(size_t)b * 3 + 2] = acc2;
}

// ---------------------------------------------------------------------------
extern "C" void kernel_launch(void* const* d_in, const int* in_sizes, int n_in,
                              void* d_out, int out_size, void* d_ws, size_t ws_size,
                              hipStream_t stream)
{
    const float* xc = (const float*)d_in[0];
    const float* xq = (const float*)d_in[1];
    const float* W1 = (const float*)d_in[2];
    const float* b1 = (const float*)d_in[3];
    const float* qw = (const float*)d_in[4];
    const float* Wq = (const float*)d_in[5];
    const float* bq = (const float*)d_in[6];
    const float* Wf = (const float*)d_in[7];
    const float* bf = (const float*)d_in[8];

    const int B = in_sizes[1] / N_QUBITS;   // 16384

    float* p_buf = (float*)d_ws;                          // 2 x B x 64 partials
    float* q_buf = p_buf + (size_t)2 * B * NOUT1;         // B x 16
    float* out   = (float*)d_out;                         // B x 3

    // GEMM partials: 128 rows per block (4 M-chunks x 2 K-slices x 32 rows)
    gemm_partial_wmma<<<B / 128, 256, 0, stream>>>(xc, W1, p_buf, B);

    quantum_kernel<<<(B + 127) / 128, 128, 0, stream>>>(xq, qw, Wq, bq, q_buf, B);

    combine_kernel<<<(B + 255) / 256, 256, 0, stream>>>(p_buf, p_buf + (size_t)B * NOUT1,
                                                        b1, q_buf, Wf, bf, out, B);
}